// GCN3_52510270160943
// MI455X (gfx1250) — compile-verified
//
#include <hip/hip_runtime.h>
#include <math.h>

#define N_NODES 22
#define BATCH   256
#define T_IN    2000

typedef float v2f __attribute__((ext_vector_type(2)));
typedef float v8f __attribute__((ext_vector_type(8)));
typedef unsigned int v4u __attribute__((ext_vector_type(4)));
typedef int v4i __attribute__((ext_vector_type(4)));
typedef int v8i __attribute__((ext_vector_type(8)));

#if defined(__has_builtin)
#if __has_builtin(__builtin_amdgcn_wmma_f32_16x16x4_f32)
#define HAVE_WMMA_F32X4 1
#endif
#if __has_builtin(__builtin_amdgcn_tensor_load_to_lds) && \
    __has_builtin(__builtin_amdgcn_s_wait_tensorcnt)
#define HAVE_TDM 1
#endif
#endif
#ifndef HAVE_WMMA_F32X4
#define HAVE_WMMA_F32X4 0
#endif
#ifndef HAVE_TDM
#define HAVE_TDM 0
#endif

__device__ __forceinline__ v8f wmma4(v2f a, v2f b, v8f c) {
#if HAVE_WMMA_F32X4
  // D = A(16x4,f32) * B(4x16,f32) + C(16x16,f32), one instr per 4-deep k-step
  return __builtin_amdgcn_wmma_f32_16x16x4_f32(false, a, false, b, (short)0, c,
                                               false, false);
#else
  c[0] += a[0] * b[0];
  return c;
#endif
}

// ---------------------------------------------------------------------------
// Tensor Data Mover: DMA `nelem` contiguous f32 from global into LDS.
// D# group0: count=1 | lds_addr | global_addr[56:0] | type=2.
// D# group1: data_size=4B, 1-D tile (tile_dim0 = tensor_dim0 = nelem).
// Issue from one wave; TDM ignores EXEC. Tracked by TENSORcnt.
// This toolchain's builtin takes 6 args (g0 v4u, g1 v8i, g2 v4i, g3 v4i,
// extra v8i, cpol i32) -- groups 2/3/extra are unused for a 1-D tile.
// ---------------------------------------------------------------------------
__device__ __forceinline__ void tdm_load_row_to_lds(const float* gsrc,
                                                    void* lds_dst,
                                                    unsigned nelem) {
#if HAVE_TDM
  unsigned long long ga = (unsigned long long)gsrc;
  v4u g0;
  g0.x = 0x1u;                                      // count=1, user D#
  g0.y = (unsigned)(unsigned long long)lds_dst;     // LDS byte address
  g0.z = (unsigned)(ga & 0xFFFFFFFFull);            // global_addr[31:0]
  g0.w = ((unsigned)(ga >> 32) & 0x01FFFFFFu) | (2u << 30);  // [56:32]|type=2
  v8i g1;
  g1[0] = (int)(2u << 16);            // workgroup_mask=0, data_size=2 (4B)
  g1[1] = (int)(nelem << 16);         // tensor_dim0[15:0] (bits 63:48)
  g1[2] = (int)((nelem >> 16) | (1u << 16));  // tensor_dim0 hi | tensor_dim1=1
  g1[3] = (int)(nelem << 16);         // tensor_dim1 hi=0 | tile_dim0
  g1[4] = 1;                          // tile_dim1=1, tile_dim2=0
  g1[5] = (int)nelem;                 // tensor_dim0_stride[31:0]
  g1[6] = 0;
  g1[7] = 0;
  v4i gz4 = {0, 0, 0, 0};
  v8i gz8 = {0, 0, 0, 0, 0, 0, 0, 0};
  __builtin_amdgcn_tensor_load_to_lds(g0, g1, gz4, gz4, gz8, 0);
#else
  (void)gsrc; (void)lds_dst; (void)nelem;
#endif
}

// ---------------------------------------------------------------------------
// Deterministic block reduction: wave32 butterfly + fixed-order cross-wave sum.
// ---------------------------------------------------------------------------
template <int V>
__device__ __forceinline__ float block_reduce_v(float* vals, float* wred) {
  const int lane = threadIdx.x & 31;
  const int wave = threadIdx.x >> 5;
#pragma unroll
  for (int m = 16; m >= 1; m >>= 1) {
#pragma unroll
    for (int v = 0; v < V; v++) vals[v] += __shfl_xor(vals[v], m, 32);
  }
  if (lane == 0) {
#pragma unroll
    for (int v = 0; v < V; v++) wred[wave * V + v] = vals[v];
  }
  __syncthreads();
  float r = 0.f;
  if (threadIdx.x < V) {
#pragma unroll
    for (int w = 0; w < 8; w++) r += wred[w * V + threadIdx.x];
  }
  return r;
}

template <int V>
__device__ __forceinline__ void block_reduce_store(float* vals, float* outp,
                                                   float* wred) {
  float r = block_reduce_v<V>(vals, wred);
  if (threadIdx.x < V) outp[threadIdx.x] = r;
}

// ---------------------------------------------------------------------------
// Fused conv stack. One block per (batch,node) row; bid = b*22 + n.
// STAGE 1: conv1 -> raw-output stats                (partial stride 8)
// STAGE 2: conv1,bn1,pool -> conv2 stats            (partial stride 16)
// STAGE 3: ... conv2,bn2,pool -> conv3 (WMMA) stats (partial stride 32)
// STAGE 4: ... conv3,bn3,relu,pool,time-mean -> feat (B,N,16)
// ---------------------------------------------------------------------------
template <int STAGE>
__global__ void __launch_bounds__(256) conv_stack(
    const float* __restrict__ x, const float* __restrict__ w1,
    const float* __restrict__ w2, const float* __restrict__ w3,
    const float* __restrict__ s1, const float* __restrict__ t1,
    const float* __restrict__ s2, const float* __restrict__ t2,
    const float* __restrict__ s3, const float* __restrict__ t3,
    float* __restrict__ partial, float* __restrict__ feat) {
  const int bid = blockIdx.x;
  const int n = bid % N_NODES;
  const int tid = threadIdx.x;

  __shared__ float xs[T_IN + 6];     // padded input row
  __shared__ float w1s[28];
  __shared__ float w2s[224];
  __shared__ float w3s[896];
  __shared__ float a1[4 * 1006];     // pooled stage-1 activations (+halo 3/3)
  __shared__ float a2[8 * 518];      // pooled stage-2 activations (+halo 3/15)
  __shared__ float wred[8 * 32];

  const float* xrow = x + (size_t)bid * T_IN;
#if HAVE_TDM
  if (tid == 0) tdm_load_row_to_lds(xrow, (void*)&xs[3], T_IN);  // async DMA
#else
  for (int j = tid; j < T_IN; j += 256) xs[3 + j] = xrow[j];
#endif
  if (tid < 3) { xs[tid] = 0.f; xs[2003 + tid] = 0.f; }
  for (int j = tid; j < 28; j += 256) w1s[j] = w1[j];
  if (STAGE >= 2) for (int j = tid; j < 224; j += 256) w2s[j] = w2[j];
  if (STAGE >= 3) for (int j = tid; j < 896; j += 256) w3s[j] = w3[j];
#if HAVE_TDM
  if (tid == 0) __builtin_amdgcn_s_wait_tensorcnt(0);
#endif
  __syncthreads();

  float w1r[4][7];
#pragma unroll
  for (int c = 0; c < 4; c++)
#pragma unroll
    for (int k = 0; k < 7; k++) w1r[c][k] = w1s[c * 7 + k];

  if (STAGE == 1) {
    float sums[8];
#pragma unroll
    for (int j = 0; j < 8; j++) sums[j] = 0.f;
    for (int i = 0; i < 8; i++) {
      int t = tid + i * 256;
      if (t < T_IN) {
#pragma unroll
        for (int c = 0; c < 4; c++) {
          float cv = 0.f;
#pragma unroll
          for (int k = 0; k < 7; k++) cv = fmaf(w1r[c][k], xs[t + k], cv);
          sums[c] += cv;
          sums[4 + c] += cv * cv;
        }
      }
    }
    block_reduce_store<8>(sums, partial + (size_t)bid * 8, wred);
    return;
  }

  // ---- conv1 + bn1 + relu + maxpool2 -> a1 (4 x 1000) ----
  float s1r[4], t1r[4];
#pragma unroll
  for (int c = 0; c < 4; c++) { s1r[c] = s1[n * 4 + c]; t1r[c] = t1[n * 4 + c]; }
  for (int i = 0; i < 4; i++) {
    int tp = tid + i * 256;
    if (tp < 1000) {
      int t0 = 2 * tp;
      float win[8];
#pragma unroll
      for (int k = 0; k < 8; k++) win[k] = xs[t0 + k];
#pragma unroll
      for (int c = 0; c < 4; c++) {
        float c0 = 0.f, c1 = 0.f;
#pragma unroll
        for (int k = 0; k < 7; k++) {
          c0 = fmaf(w1r[c][k], win[k], c0);
          c1 = fmaf(w1r[c][k], win[k + 1], c1);
        }
        float v0 = fmaxf(fmaf(c0, s1r[c], t1r[c]), 0.f);
        float v1 = fmaxf(fmaf(c1, s1r[c], t1r[c]), 0.f);
        a1[c * 1006 + 3 + tp] = fmaxf(v0, v1);
      }
    }
  }
  if (tid < 3) {
#pragma unroll
    for (int c = 0; c < 4; c++) {
      a1[c * 1006 + tid] = 0.f;
      a1[c * 1006 + 1003 + tid] = 0.f;
    }
  }
  __syncthreads();

  if (STAGE == 2) {
    float sums[16];
#pragma unroll
    for (int j = 0; j < 16; j++) sums[j] = 0.f;
    for (int i = 0; i < 4; i++) {
      int t = tid + i * 256;
      if (t < 1000) {
        float acc[8];
#pragma unroll
        for (int oc = 0; oc < 8; oc++) acc[oc] = 0.f;
#pragma unroll
        for (int ic = 0; ic < 4; ic++) {
          float win[7];
#pragma unroll
          for (int k = 0; k < 7; k++) win[k] = a1[ic * 1006 + t + k];
#pragma unroll
          for (int oc = 0; oc < 8; oc++) {
            float w = acc[oc];
#pragma unroll
            for (int k = 0; k < 7; k++)
              w = fmaf(w2s[oc * 28 + ic * 7 + k], win[k], w);
            acc[oc] = w;
          }
        }
#pragma unroll
        for (int oc = 0; oc < 8; oc++) {
          sums[oc] += acc[oc];
          sums[8 + oc] += acc[oc] * acc[oc];
        }
      }
    }
    block_reduce_store<16>(sums, partial + (size_t)bid * 16, wred);
    return;
  }

  // ---- conv2 + bn2 + relu + maxpool2 -> a2 (8 x 500) ----
  float s2r[8], t2r[8];
#pragma unroll
  for (int c = 0; c < 8; c++) { s2r[c] = s2[n * 8 + c]; t2r[c] = t2[n * 8 + c]; }
  for (int i = 0; i < 2; i++) {
    int tp = tid + i * 256;
    if (tp < 500) {
      int t0 = 2 * tp;
      float acc0[8], acc1[8];
#pragma unroll
      for (int oc = 0; oc < 8; oc++) { acc0[oc] = 0.f; acc1[oc] = 0.f; }
#pragma unroll
      for (int ic = 0; ic < 4; ic++) {
        float win[8];
#pragma unroll
        for (int k = 0; k < 8; k++) win[k] = a1[ic * 1006 + t0 + k];
#pragma unroll
        for (int oc = 0; oc < 8; oc++) {
          float u0 = acc0[oc], u1 = acc1[oc];
#pragma unroll
          for (int k = 0; k < 7; k++) {
            float wv = w2s[oc * 28 + ic * 7 + k];
            u0 = fmaf(wv, win[k], u0);
            u1 = fmaf(wv, win[k + 1], u1);
          }
          acc0[oc] = u0;
          acc1[oc] = u1;
        }
      }
#pragma unroll
      for (int oc = 0; oc < 8; oc++) {
        float v0 = fmaxf(fmaf(acc0[oc], s2r[oc], t2r[oc]), 0.f);
        float v1 = fmaxf(fmaf(acc1[oc], s2r[oc], t2r[oc]), 0.f);
        a2[oc * 518 + 3 + tp] = fmaxf(v0, v1);
      }
    }
  }
  if (tid < 3) {
#pragma unroll
    for (int oc = 0; oc < 8; oc++) a2[oc * 518 + tid] = 0.f;
  }
  if (tid < 15) {
#pragma unroll
    for (int oc = 0; oc < 8; oc++) a2[oc * 518 + 503 + tid] = 0.f;
  }
  __syncthreads();

  // ---- conv3 as implicit GEMM via V_WMMA_F32_16X16X4_F32 ----
  // C(16 oc x 512 t) = W3(16 x 56) @ patches(56 x 512); 8 waves x 4 tiles.
  const int lane = tid & 31;
  const int wave = tid >> 5;
  const int half = lane >> 4;
  const int l15 = lane & 15;

  if (STAGE == 3) {
    float s8[8], q8[8];
#pragma unroll
    for (int r = 0; r < 8; r++) { s8[r] = 0.f; q8[r] = 0.f; }
#pragma unroll
    for (int q = 0; q < 4; q++) {
      const int nt = wave * 4 + q;
      const int t = nt * 16 + l15;
      v8f acc = {};
#pragma unroll
      for (int k0 = 0; k0 < 56; k0 += 4) {
        int ka = k0 + half * 2;
        v2f a = {w3s[l15 * 56 + ka], w3s[l15 * 56 + ka + 1]};
        v2f b = {a2[(ka / 7) * 518 + t + (ka % 7)],
                 a2[((ka + 1) / 7) * 518 + t + ((ka + 1) % 7)]};
        acc = wmma4(a, b, acc);
      }
      const float valid = (t < 500) ? 1.f : 0.f;
#pragma unroll
      for (int r = 0; r < 8; r++) {
        float v = acc[r] * valid;
        s8[r] += v;
        q8[r] += v * v;
      }
    }
#pragma unroll
    for (int m = 8; m >= 1; m >>= 1) {
#pragma unroll
      for (int r = 0; r < 8; r++) {
        s8[r] += __shfl_xor(s8[r], m, 32);
        q8[r] += __shfl_xor(q8[r], m, 32);
      }
    }
    if (l15 == 0) {
#pragma unroll
      for (int r = 0; r < 8; r++) {
        wred[wave * 32 + half * 8 + r] = s8[r];
        wred[wave * 32 + 16 + half * 8 + r] = q8[r];
      }
    }
    __syncthreads();
    if (tid < 32) {
      float r = 0.f;
#pragma unroll
      for (int w = 0; w < 8; w++) r += wred[w * 32 + tid];
      partial[(size_t)bid * 32 + tid] = r;
    }
    return;
  }

  // STAGE 4: bn3 + relu + pairwise maxpool + mean over 250 -> feat
  float s3r[8], t3r[8];
#pragma unroll
  for (int r = 0; r < 8; r++) {
    int ch = r + half * 8;
    s3r[r] = s3[n * 16 + ch];
    t3r[r] = t3[n * 16 + ch];
  }
  float f8[8];
#pragma unroll
  for (int r = 0; r < 8; r++) f8[r] = 0.f;
#pragma unroll
  for (int q = 0; q < 4; q++) {
    const int nt = wave * 4 + q;
    const int t = nt * 16 + l15;
    v8f acc = {};
#pragma unroll
    for (int k0 = 0; k0 < 56; k0 += 4) {
      int ka = k0 + half * 2;
      v2f a = {w3s[l15 * 56 + ka], w3s[l15 * 56 + ka + 1]};
      v2f b = {a2[(ka / 7) * 518 + t + (ka % 7)],
               a2[((ka + 1) / 7) * 518 + t + ((ka + 1) % 7)]};
      acc = wmma4(a, b, acc);
    }
    const bool contrib = ((l15 & 1) == 0) && (t < 500);
#pragma unroll
    for (int r = 0; r < 8; r++) {
      float v = fmaxf(fmaf(acc[r], s3r[r], t3r[r]), 0.f);
      float vn = __shfl_xor(v, 1, 32);     // partner column of the pool pair
      float pm = fmaxf(v, vn);
      if (contrib) f8[r] += pm;
    }
  }
#pragma unroll
  for (int m = 8; m >= 1; m >>= 1) {
#pragma unroll
    for (int r = 0; r < 8; r++) f8[r] += __shfl_xor(f8[r], m, 32);
  }
  if (l15 == 0) {
#pragma unroll
    for (int r = 0; r < 8; r++) wred[wave * 16 + half * 8 + r] = f8[r];
  }
  __syncthreads();
  if (tid < 16) {
    float r = 0.f;
#pragma unroll
    for (int w = 0; w < 8; w++) r += wred[w * 16 + tid];
    feat[(size_t)bid * 16 + tid] = r * (1.0f / 250.0f);
  }
}

// ---------------------------------------------------------------------------
// Per-node BN stats finalize: sum 256 batch partials in fixed order, emit
// scale s = gamma * rsqrt(var+eps) and shift t = beta - mean*s.
// ---------------------------------------------------------------------------
template <int C>
__global__ void __launch_bounds__(256) finalize_stats(
    const float* __restrict__ partial, const float* __restrict__ gamma,
    const float* __restrict__ beta, float* __restrict__ s,
    float* __restrict__ t, float invCnt) {
  constexpr int V = 2 * C;
  const int n = blockIdx.x;
  float vals[V];
  const float* p = partial + ((size_t)threadIdx.x * N_NODES + n) * V;
#pragma unroll
  for (int j = 0; j < V; j++) vals[j] = p[j];
  __shared__ float wred[8 * 32];
  __shared__ float res[32];
  float r = block_reduce_v<V>(vals, wred);
  if (threadIdx.x < V) res[threadIdx.x] = r;
  __syncthreads();
  if (threadIdx.x < C) {
    const int c = threadIdx.x;
    float mean = res[c] * invCnt;
    float var = res[C + c] * invCnt - mean * mean;
    float sv = gamma[c] * rsqrtf(var + 1e-5f);
    s[n * C + c] = sv;
    t[n * C + c] = beta[c] - mean * sv;
  }
}

// ---------------------------------------------------------------------------
// Single-wave f32 WMMA GEMM over zero-padded tiles. Rows >= Mvalid are
// stored as 0 (keeps padding invariant, no EXEC manipulation in store loop).
// C must have Mtiles*16 rows allocated.
// ---------------------------------------------------------------------------
__device__ __forceinline__ void wave_gemm(const float* A, int lda,
                                          const float* B, int ldb, float* C,
                                          int ldc, int Mtiles, int Ntiles,
                                          int K, const float* bias, bool relu,
                                          int Mvalid) {
  const int lane = threadIdx.x & 31;
  const int half = lane >> 4;
  const int l15 = lane & 15;
  for (int mt = 0; mt < Mtiles; mt++) {
    for (int nt = 0; nt < Ntiles; nt++) {
      v8f acc = {};
      const int col = nt * 16 + l15;
      for (int k0 = 0; k0 < K; k0 += 4) {
        int ka = k0 + half * 2;
        v2f a = {A[(mt * 16 + l15) * lda + ka],
                 A[(mt * 16 + l15) * lda + ka + 1]};
        v2f b = {B[ka * ldb + col], B[(ka + 1) * ldb + col]};
        acc = wmma4(a, b, acc);
      }
      float bv = (bias != nullptr) ? bias[col] : 0.f;
#pragma unroll
      for (int r = 0; r < 8; r++) {
        int row = mt * 16 + r + half * 8;
        float v = acc[r];
        if (bias != nullptr) v += bv;
        if (relu) v = fmaxf(v, 0.f);
        v = (row < Mvalid) ? v : 0.f;   // select, not branch
        C[row * ldc + col] = v;
      }
    }
  }
}

// ---------------------------------------------------------------------------
// Per-sample graph head: adjacency (top-4 NN) + 3 GCN layers (WMMA) + MLP.
// One wave per block (WMMA needs full-EXEC wave32).
// ---------------------------------------------------------------------------
__global__ void __launch_bounds__(32) gcn_head(
    const float* __restrict__ feat, const float* __restrict__ gw1,
    const float* __restrict__ gb1, const float* __restrict__ gw2,
    const float* __restrict__ gb2, const float* __restrict__ gw3,
    const float* __restrict__ gb3, const float* __restrict__ fw1,
    const float* __restrict__ fb1, const float* __restrict__ fw2,
    const float* __restrict__ fb2, float* __restrict__ out) {
  const int b = blockIdx.x;
  const int lane = threadIdx.x;

  __shared__ float X[32 * 128];    // node features, zero-padded to 32 rows
  __shared__ float Z[32 * 128];    // projected features, rows >=22 zero
  __shared__ float adj[32 * 24];   // adjacency, zero-padded
  __shared__ float dist[22 * 22];
  __shared__ float fs[22 * 16];
  __shared__ float pooled[128];
  __shared__ float h1[64];

  for (int i = lane; i < 22 * 16; i += 32) fs[i] = feat[(size_t)b * 22 * 16 + i];
  for (int i = lane; i < 32 * 128; i += 32) X[i] = 0.f;
  for (int i = lane; i < 32 * 24; i += 32) adj[i] = 0.f;
  __syncthreads();

  for (int idx = lane; idx < 484; idx += 32) {
    int i = idx / 22, j = idx % 22;
    float d = 0.f;
#pragma unroll
    for (int c = 0; c < 16; c++) {
      float df = fs[i * 16 + c] - fs[j * 16 + c];
      d = fmaf(df, df, d);
    }
    dist[idx] = d;
  }
  __syncthreads();

  if (lane < 22) {  // top-4 smallest (self dist == 0 -> always included)
    for (int it = 0; it < 4; it++) {
      float best = 3.0e38f;
      int bj = 0;
      for (int j = 0; j < 22; j++) {
        float v = dist[lane * 22 + j];
        if (v < best) { best = v; bj = j; }
      }
      adj[lane * 24 + bj] = 0.25f;
      dist[lane * 22 + bj] = 3.0e38f;
    }
  }
  for (int idx = lane; idx < 22 * 16; idx += 32)
    X[(idx / 16) * 128 + (idx % 16)] = fs[idx];
  __syncthreads();

  // layer 1: Z = X @ gw1 (22x16 @ 16x32) ; X = relu(adj @ Z + gb1)
  wave_gemm(X, 128, gw1, 32, Z, 128, 2, 2, 16, nullptr, false, 22);
  __syncthreads();
  wave_gemm(adj, 24, Z, 128, X, 128, 2, 2, 24, gb1, true, 22);
  __syncthreads();
  // layer 2: 22x32 @ 32x64
  wave_gemm(X, 128, gw2, 64, Z, 128, 2, 4, 32, nullptr, false, 22);
  __syncthreads();
  wave_gemm(adj, 24, Z, 128, X, 128, 2, 4, 24, gb2, true, 22);
  __syncthreads();
  // layer 3: 22x64 @ 64x128
  wave_gemm(X, 128, gw3, 128, Z, 128, 2, 8, 64, nullptr, false, 22);
  __syncthreads();
  wave_gemm(adj, 24, Z, 128, X, 128, 2, 8, 24, gb3, true, 22);
  __syncthreads();

  for (int c = lane; c < 128; c += 32) {
    float s = 0.f;
#pragma unroll
    for (int r = 0; r < 22; r++) s += X[r * 128 + c];
    pooled[c] = s * (1.0f / 22.0f);
  }
  __syncthreads();
  for (int o = lane; o < 64; o += 32) {
    float s = fb1[o];
#pragma unroll 8
    for (int c = 0; c < 128; c++) s = fmaf(pooled[c], fw1[c * 64 + o], s);
    h1[o] = s;
  }
  __syncthreads();
  if (lane < 4) {
    float s = fb2[lane];
#pragma unroll
    for (int o = 0; o < 64; o++) s = fmaf(h1[o], fw2[o * 4 + lane], s);
    out[(size_t)b * 4 + lane] = s;
  }
}

// ---------------------------------------------------------------------------
extern "C" void kernel_launch(void* const* d_in, const int* in_sizes, int n_in,
                              void* d_out, int out_size, void* d_ws,
                              size_t ws_size, hipStream_t stream) {
  (void)in_sizes; (void)n_in; (void)out_size; (void)ws_size;
  const float* x   = (const float*)d_in[0];
  const float* w1  = (const float*)d_in[1];
  const float* g1  = (const float*)d_in[2];
  const float* b1  = (const float*)d_in[3];
  const float* w2  = (const float*)d_in[4];
  const float* g2  = (const float*)d_in[5];
  const float* b2  = (const float*)d_in[6];
  const float* w3  = (const float*)d_in[7];
  const float* g3  = (const float*)d_in[8];
  const float* b3  = (const float*)d_in[9];
  const float* gw1 = (const float*)d_in[10];
  const float* gb1 = (const float*)d_in[11];
  const float* gw2 = (const float*)d_in[12];
  const float* gb2 = (const float*)d_in[13];
  const float* gw3 = (const float*)d_in[14];
  const float* gb3 = (const float*)d_in[15];
  const float* fw1 = (const float*)d_in[16];
  const float* fb1 = (const float*)d_in[17];
  const float* fw2 = (const float*)d_in[18];
  const float* fb2 = (const float*)d_in[19];
  float* out = (float*)d_out;

  const int NB = BATCH * N_NODES;  // 5632 (b*22+n) rows
  float* ws = (float*)d_ws;
  float* partial = ws;                       // NB * 32
  float* s1 = ws + (size_t)NB * 32;          // 22*4 each
  float* t1 = s1 + N_NODES * 4;
  float* s2 = t1 + N_NODES * 4;              // 22*8 each
  float* t2 = s2 + N_NODES * 8;
  float* s3 = t2 + N_NODES * 8;              // 22*16 each
  float* t3 = s3 + N_NODES * 16;
  float* feat = t3 + N_NODES * 16;           // 256*22*16

  dim3 blk(256);
  conv_stack<1><<<NB, blk, 0, stream>>>(x, w1, w2, w3, s1, t1, s2, t2, s3, t3,
                                        partial, feat);
  finalize_stats<4><<<N_NODES, blk, 0, stream>>>(partial, g1, b1, s1, t1,
                                                 1.0f / (256.0f * 2000.0f));
  conv_stack<2><<<NB, blk, 0, stream>>>(x, w1, w2, w3, s1, t1, s2, t2, s3, t3,
                                        partial, feat);
  finalize_stats<8><<<N_NODES, blk, 0, stream>>>(partial, g2, b2, s2, t2,
                                                 1.0f / (256.0f * 1000.0f));
  conv_stack<3><<<NB, blk, 0, stream>>>(x, w1, w2, w3, s1, t1, s2, t2, s3, t3,
                                        partial, feat);
  finalize_stats<16><<<N_NODES, blk, 0, stream>>>(partial, g3, b3, s3, t3,
                                                  1.0f / (256.0f * 500.0f));
  conv_stack<4><<<NB, blk, 0, stream>>>(x, w1, w2, w3, s1, t1, s2, t2, s3, t3,
                                        partial, feat);
  gcn_head<<<BATCH, dim3(32), 0, stream>>>(feat, gw1, gb1, gw2, gb2, gw3, gb3,
                                           fw1, fb1, fw2, fb2, out);
}